// SparseLinear_71511205479093
// MI455X (gfx1250) — compile-verified
//
#include <hip/hip_runtime.h>
#include <stdint.h>

// Problem constants (fixed by the reference).
#define M_ROWS    8192
#define N_IN      32768
#define N_OUT     1024
#define TILE_ROWS 16
#define NUM_TILES (M_ROWS / TILE_ROWS)   // 512
#define CAP       1024                   // padded nnz capacity per 16-row tile (mean ~512, max ~600)
#define OVF_CAP   4096
#define CHUNK     512                    // nnz per packing chunk

typedef float v2f __attribute__((ext_vector_type(2)));
typedef float v8f __attribute__((ext_vector_type(8)));

// ---------------------------------------------------------------------------
// Phase A: per-chunk, per-tile histogram of nnz counts (deterministic counts).
// ---------------------------------------------------------------------------
__global__ void __launch_bounds__(256)
k_hist(const int* __restrict__ rows, int nnz, uint32_t* __restrict__ hist) {
    __shared__ uint32_t h[NUM_TILES];
    const int tid = threadIdx.x;
    const int chunk = blockIdx.x;
    for (int i = tid; i < NUM_TILES; i += 256) h[i] = 0u;
    __syncthreads();
    const int k0 = chunk * CHUNK;
    const int cn = min(CHUNK, nnz - k0);
    for (int i = tid; i < cn; i += 256)
        atomicAdd(&h[(uint32_t)rows[k0 + i] >> 4], 1u);
    __syncthreads();
    for (int i = tid; i < NUM_TILES; i += 256)
        hist[(size_t)chunk * NUM_TILES + i] = h[i];
}

// ---------------------------------------------------------------------------
// Phase B: serial exclusive scan over chunks, per tile (deterministic bases).
// ---------------------------------------------------------------------------
__global__ void __launch_bounds__(256)
k_scan(uint32_t* __restrict__ hist, int nchunk,
       uint32_t* __restrict__ counts, uint32_t* __restrict__ ovf_count) {
    const int t = blockIdx.x * blockDim.x + threadIdx.x;
    if (t == 0) *ovf_count = 0u;
    if (t >= NUM_TILES) return;
    uint32_t run = 0u;
    for (int c = 0; c < nchunk; ++c) {
        uint32_t v = hist[(size_t)c * NUM_TILES + t];
        hist[(size_t)c * NUM_TILES + t] = run;
        run += v;
    }
    counts[t] = run;
}

// ---------------------------------------------------------------------------
// Phase C: deterministic in-chunk rank assignment, scatter into ELL tiles.
// Entry = uint2 { f32 value bits, meta = col[15:0] | rowInTile[19:16] }
// ---------------------------------------------------------------------------
__global__ void __launch_bounds__(256)
k_pack(const int* __restrict__ rows, const int* __restrict__ cols,
       const float* __restrict__ values, int nnz,
       const uint32_t* __restrict__ hist,
       uint2* __restrict__ tent,
       uint32_t* __restrict__ ovf_count, uint32_t* __restrict__ ovf_list) {
    __shared__ uint32_t base[NUM_TILES];
    __shared__ uint32_t slot[CHUNK];
    __shared__ int      srow[CHUNK];
    const int tid = threadIdx.x;
    const int chunk = blockIdx.x;
    const int k0 = chunk * CHUNK;
    const int cn = min(CHUNK, nnz - k0);
    for (int i = tid; i < NUM_TILES; i += 256)
        base[i] = hist[(size_t)chunk * NUM_TILES + i];
    for (int i = tid; i < cn; i += 256)
        srow[i] = rows[k0 + i];
    __syncthreads();
    if (tid == 0) {                       // serial, order = input order (deterministic)
        for (int i = 0; i < cn; ++i) {
            int t = (uint32_t)srow[i] >> 4;
            slot[i] = base[t]++;
        }
    }
    __syncthreads();
    for (int i = tid; i < cn; i += 256) {
        const int r = srow[i];
        const int t = (uint32_t)r >> 4;
        const uint32_t s = slot[i];
        if (s < CAP) {
            uint2 e;
            e.x = __float_as_uint(values[k0 + i]);
            e.y = ((uint32_t)cols[k0 + i] & 0xFFFFu) | (((uint32_t)r & 15u) << 16);
            tent[(size_t)t * CAP + s] = e;
        } else {
            uint32_t o = atomicAdd(ovf_count, 1u);
            if (o < OVF_CAP) ovf_list[o] = (uint32_t)(k0 + i);
        }
    }
}

// ---------------------------------------------------------------------------
// Main SpMM via V_WMMA_F32_16X16X4_F32.
// Block = 256 threads = 8 waves (wave32); each wave owns a 16x16 output tile.
// grid.x = row tile (512), grid.y = 128-column slab (8).
// A[16x4]: lanes 0-15 hold K=0,1 in v[0],v[1]; lanes 16-31 hold K=2,3.
// B[4x16]: VGPR0 = K0 (lanes 0-15) / K2 (lanes 16-31); VGPR1 = K1/K3.
// C/D[16x16]: VGPR v -> M=v (lanes 0-15) / M=v+8 (lanes 16-31), N=lane&15.
// Inner loop (software-pipelined): one ds_load_b128 (two packed entries for
// this lane-half), two coalesced 64B global loads of L2-resident weight rows,
// two prefetches of the next chunk's rows, one WMMA. A-selects are cndmask
// (values loaded unconditionally), no divergent branches anywhere in the loop.
// ---------------------------------------------------------------------------
__global__ void __launch_bounds__(256)
k_spmm_wmma(const float* __restrict__ W, const float* __restrict__ bias,
            const uint32_t* __restrict__ counts,
            const uint2* __restrict__ tent, float* __restrict__ out) {
    __shared__ uint4 s_ent4[CAP / 2];           // pairs of {val,meta} entries, 16B aligned
    const int tile = blockIdx.x;
    const int tid  = threadIdx.x;

    uint32_t cnt = counts[tile];
    if (cnt > CAP) cnt = CAP;
    const uint32_t kpad = (cnt + 3u) & ~3u;

    {   // cooperative fill; padded slots are exact zeros (val=0 => A element 0)
        uint2* s_ent2 = (uint2*)s_ent4;
        for (uint32_t i = tid; i < kpad; i += 256) {
            uint2 e = (i < cnt) ? tent[(size_t)tile * CAP + i] : make_uint2(0u, 0u);
            s_ent2[i] = e;
        }
    }
    __syncthreads();

    const int wave = tid >> 5;          // wave32
    const int lane = tid & 31;
    const int half = lane >> 4;         // 0: K pair (0,1), 1: K pair (2,3)
    const int l    = lane & 15;         // M (for A) / N (for B,C,D) within tile
    const int nc   = blockIdx.y * 128 + wave * 16 + l;   // global output column
    const float* __restrict__ Wc = W + nc;               // column-offset base
    const float bv = bias[nc];

    v8f acc = {0.f, 0.f, 0.f, 0.f, 0.f, 0.f, 0.f, 0.f};

    if (kpad) {
        uint4 e = s_ent4[half];                          // entries (2*half, 2*half+1)
        for (uint32_t j = 0; j < kpad; j += 4) {
            // --- branchless pipelined fetch of next chunk's entry pair ---
            const uint32_t jn = (j + 4 < kpad) ? (j + 4) : j;   // uniform s_cselect
            const uint4 en = s_ent4[(jn >> 1) + (uint32_t)half];
            __builtin_prefetch(&Wc[(size_t)(en.y & 0xFFFFu) * N_OUT], 0, 1);
            __builtin_prefetch(&Wc[(size_t)(en.w & 0xFFFFu) * N_OUT], 0, 1);

            // --- operands for this chunk ---
            const float    v0 = __uint_as_float(e.x);
            const uint32_t m0 = e.y;
            const float    v1 = __uint_as_float(e.z);
            const uint32_t m1 = e.w;
            v2f a, b;
            a.x = ((m0 >> 16) == (uint32_t)l) ? v0 : 0.0f;      // v_cndmask
            a.y = ((m1 >> 16) == (uint32_t)l) ? v1 : 0.0f;
            b.x = Wc[(size_t)(m0 & 0xFFFFu) * N_OUT];           // 64B coalesced, L2-resident
            b.y = Wc[(size_t)(m1 & 0xFFFFu) * N_OUT];

            acc = __builtin_amdgcn_wmma_f32_16x16x4_f32(
                /*neg_a=*/false, a, /*neg_b=*/false, b,
                /*c_mod=*/(short)0, acc, /*reuse_a=*/false, /*reuse_b=*/false);
            e = en;
        }
    }

    const int rbase = tile * TILE_ROWS + half * 8;
#pragma unroll
    for (int v = 0; v < 8; ++v)
        out[(size_t)(rbase + v) * N_OUT + nc] = acc[v] + bv;
}

// ---------------------------------------------------------------------------
// Overflow fixup (never triggers for this distribution; correctness guard).
// ---------------------------------------------------------------------------
__global__ void __launch_bounds__(256)
k_fixup(const int* __restrict__ rows, const int* __restrict__ cols,
        const float* __restrict__ values, const float* __restrict__ W,
        const uint32_t* __restrict__ ovf_count,
        const uint32_t* __restrict__ ovf_list, float* __restrict__ out) {
    uint32_t n = *ovf_count;
    if (n > OVF_CAP) n = OVF_CAP;
    for (uint32_t e = blockIdx.x; e < n; e += gridDim.x) {
        const uint32_t k = ovf_list[e];
        const int r = rows[k];
        const int c = cols[k];
        const float v = values[k];
        for (int o = threadIdx.x; o < N_OUT; o += 256)
            atomicAdd(&out[(size_t)r * N_OUT + o], v * W[(size_t)c * N_OUT + o]);
    }
}

// ---------------------------------------------------------------------------
extern "C" void kernel_launch(void* const* d_in, const int* in_sizes, int n_in,
                              void* d_out, int out_size, void* d_ws, size_t ws_size,
                              hipStream_t stream) {
    const int    nnz     = in_sizes[1];
    const int*   rows    = (const int*)d_in[0];
    const int*   cols    = rows + nnz;                // indices is [2, nnz]
    const float* values  = (const float*)d_in[1];
    const float* W       = (const float*)d_in[4];
    const float* bias    = (const float*)d_in[5];
    float*       out     = (float*)d_out;

    const int nchunk = (nnz + CHUNK - 1) / CHUNK;

    // Workspace layout (~5.3 MB total)
    char* ws = (char*)d_ws;
    uint32_t* counts    = (uint32_t*)(ws + 0);              // 2 KB
    uint32_t* ovf_count = (uint32_t*)(ws + 2048);           // 4 B
    uint32_t* ovf_list  = (uint32_t*)(ws + 4096);           // 16 KB
    uint32_t* hist      = (uint32_t*)(ws + 32768);          // nchunk*512*4 (1 MB)
    size_t hist_bytes   = ((size_t)nchunk * NUM_TILES * 4 + 255) & ~(size_t)255;
    uint2*    tent      = (uint2*)(ws + 32768 + hist_bytes);                    // 4 MB

    k_hist<<<nchunk, 256, 0, stream>>>(rows, nnz, hist);
    k_scan<<<2, 256, 0, stream>>>(hist, nchunk, counts, ovf_count);
    k_pack<<<nchunk, 256, 0, stream>>>(rows, cols, values, nnz, hist,
                                       tent, ovf_count, ovf_list);
    dim3 grid(NUM_TILES, N_OUT / 128);
    k_spmm_wmma<<<grid, 256, 0, stream>>>(W, bias, counts, tent, out);
    k_fixup<<<64, 256, 0, stream>>>(rows, cols, values, W, ovf_count, ovf_list, out);
}